// GCN_10050223473071
// MI455X (gfx1250) — compile-verified
//
#include <hip/hip_runtime.h>
#include <hip/hip_bf16.h>

typedef __attribute__((ext_vector_type(2))) float v2f;
typedef __attribute__((ext_vector_type(8))) float v8f;

#define N_NODES   50000
#define N_EDGES   1600000
#define D_FEAT    512
#define D_H1      128
#define D_H2      32
#define NEG_SLOPE 0.01f

#define M_TILES   3125           // 50000 / 16, exact
#define WPB       8              // waves per block in GEMM kernels

// Guaranteed-native fp32 atomic add: the agg buffers live in plain device HBM
// (no fine-grained / remote / denorm concerns), so tell clang it may lower
// atomicrmw fadd to global_atomic_add_f32 instead of a CAS loop.
__device__ __forceinline__ void atomic_add_f32(float* p, float v) {
    [[clang::atomic(no_remote_memory, no_fine_grained_memory, ignore_denormal_mode)]]
    {
        __hip_atomic_fetch_add(p, v, __ATOMIC_RELAXED, __HIP_MEMORY_SCOPE_AGENT);
    }
}

// ---------------- degree / normalization ----------------

__global__ void k_deg_init(float* __restrict__ deg) {
    int i = blockIdx.x * blockDim.x + threadIdx.x;
    if (i < N_NODES) deg[i] = 1.0f;       // self-loop contribution
}

__global__ void k_deg_accum(const long long* __restrict__ col, float* __restrict__ deg) {
    int e = blockIdx.x * blockDim.x + threadIdx.x;
    if (e < N_EDGES) atomic_add_f32(&deg[(int)col[e]], 1.0f);
}

__global__ void k_dinv(float* __restrict__ deg) {
    int i = blockIdx.x * blockDim.x + threadIdx.x;
    if (i < N_NODES) deg[i] = rsqrtf(deg[i]);   // deg >= 1 always (self loops)
}

// ---------------- fp32 WMMA GEMM: H[M,NOUT] = A[M,K] @ W[K,NOUT] ----------------
// One 16x16 output tile per wave, K consumed 4 at a time via v_wmma_f32_16x16x4_f32.
// W's 16-column slab for this blockIdx.y is staged in LDS.

template<int K, int NOUT>
__global__ void k_gemm_wmma(const float* __restrict__ A,
                            const float* __restrict__ W,
                            float* __restrict__ H) {
    __shared__ float Wcol[K * 16];
    const int ntile = blockIdx.y;
    for (int i = threadIdx.x; i < K * 16; i += blockDim.x) {
        int k = i >> 4, n = i & 15;
        Wcol[i] = W[k * NOUT + ntile * 16 + n];
    }
    __syncthreads();

    const int wave  = threadIdx.x >> 5;
    const int lane  = threadIdx.x & 31;
    const int mtile = blockIdx.x * WPB + wave;
    if (mtile >= M_TILES) return;          // wave-uniform: EXEC stays all-1s for WMMA

    const int mrow  = lane & 15;
    const int khalf = lane >> 4;           // 0: K pair {0,1}; 1: K pair {2,3}
    const int n     = lane & 15;

    const float* arow = A + (size_t)(mtile * 16 + mrow) * K + 2 * khalf;

    v8f c = {};
#pragma unroll 4
    for (int k = 0; k < K; k += 4) {
        v2f a = *(const v2f*)(arow + k);                       // A[m][k+2h .. k+2h+1]
        v2f b;
        b.x = Wcol[(k + 2 * khalf)     * 16 + n];              // B[k+2h][n]
        b.y = Wcol[(k + 2 * khalf + 1) * 16 + n];              // B[k+2h+1][n]
        c = __builtin_amdgcn_wmma_f32_16x16x4_f32(
                /*neg_a=*/false, a, /*neg_b=*/false, b,
                /*c_mod=*/(short)0, c, /*reuse_a=*/false, /*reuse_b=*/false);
    }

    // D layout: VGPR r -> (M = r + 8*khalf, N = lane&15)
    float* hrow = H + (size_t)(mtile * 16 + 8 * khalf) * NOUT + ntile * 16 + n;
#pragma unroll
    for (int r = 0; r < 8; ++r) hrow[(size_t)r * NOUT] = c[r];
}

// ---------------- edge scatter: agg[col] += h[row] * dinv[row]*dinv[col] ----------------
// Each thread handles one edge x 4 features (float4 gather + 4 hw fp atomics).

template<int F>
__global__ void k_scatter(const long long* __restrict__ row,
                          const long long* __restrict__ col,
                          const float* __restrict__ dinv,
                          const float* __restrict__ h,
                          float* __restrict__ agg) {
    const int per_edge = F / 4;
    long long gid = (long long)blockIdx.x * blockDim.x + threadIdx.x;
    long long e   = gid / per_edge;
    int       q   = (int)(gid - e * per_edge);
    if (e >= N_EDGES) return;
    int r = (int)row[e];
    int c = (int)col[e];
    float w = dinv[r] * dinv[c];
    const float4 v = *(const float4*)(h + (size_t)r * F + q * 4);
    float* dst = agg + (size_t)c * F + q * 4;
    atomic_add_f32(dst + 0, v.x * w);
    atomic_add_f32(dst + 1, v.y * w);
    atomic_add_f32(dst + 2, v.z * w);
    atomic_add_f32(dst + 3, v.w * w);
}

// ---------------- post: y = leaky(agg + h*dinv^2 (self loop) + bias), in place ----------------

template<int F>
__global__ void k_post(const float* __restrict__ h,
                       const float* __restrict__ dinv,
                       const float* __restrict__ bias,
                       float* __restrict__ agg) {
    long long i = (long long)blockIdx.x * blockDim.x + threadIdx.x;
    if (i >= (long long)N_NODES * F) return;
    int node = (int)(i / F);
    int f    = (int)(i - (long long)node * F);
    float d = dinv[node];
    float y = agg[i] + h[i] * d * d + bias[f];
    agg[i] = (y >= 0.0f) ? y : NEG_SLOPE * y;
}

// ---------------- final: softmax(z @ Wl + bl) over 2 classes ----------------

__global__ void k_final(const float* __restrict__ z,
                        const float* __restrict__ Wl,
                        const float* __restrict__ bl,
                        float* __restrict__ out) {
    int i = blockIdx.x * blockDim.x + threadIdx.x;
    if (i >= N_NODES) return;
    const float* zr = z + (size_t)i * D_H2;
    float a0 = bl[0], a1 = bl[1];
#pragma unroll
    for (int f = 0; f < D_H2; ++f) {
        float v = zr[f];
        a0 += v * Wl[f * 2 + 0];
        a1 += v * Wl[f * 2 + 1];
    }
    float m  = fmaxf(a0, a1);
    float e0 = __expf(a0 - m);
    float e1 = __expf(a1 - m);
    float s  = e0 + e1;
    out[(size_t)i * 2 + 0] = e0 / s;
    out[(size_t)i * 2 + 1] = e1 / s;
}

// ---------------- launch ----------------

extern "C" void kernel_launch(void* const* d_in, const int* in_sizes, int n_in,
                              void* d_out, int out_size, void* d_ws, size_t ws_size,
                              hipStream_t stream) {
    const float*     x   = (const float*)d_in[0];
    const long long* ei  = (const long long*)d_in[1];   // int64 [2, N_EDGES]
    const float*     W1  = (const float*)d_in[2];
    const float*     b1  = (const float*)d_in[3];
    const float*     W2  = (const float*)d_in[4];
    const float*     b2  = (const float*)d_in[5];
    const float*     Wl  = (const float*)d_in[6];
    const float*     bl  = (const float*)d_in[7];
    float*           out = (float*)d_out;

    const long long* row = ei;
    const long long* col = ei + N_EDGES;

    // Workspace layout (floats):
    //   dinv : [0, 50000)
    //   h1   : [51200, 51200+6.4M)   -- reused: h2 = first 1.6M, agg2 = next 1.6M (h1 dead by then)
    //   agg1 : [51200+6.4M, +6.4M)   -- becomes y1 in place
    float* ws   = (float*)d_ws;
    float* dinv = ws;
    float* h1   = ws + 51200;
    float* agg1 = ws + 51200 + 6400000;
    float* h2   = h1;
    float* agg2 = h1 + 1600000;

    const int T = 256;

    // zero agg1 for this launch (deterministic across graph replays)
    hipMemsetAsync(agg1, 0, (size_t)N_NODES * D_H1 * sizeof(float), stream);

    // degree -> dinv
    k_deg_init <<<(N_NODES + T - 1) / T, T, 0, stream>>>(dinv);
    k_deg_accum<<<(N_EDGES + T - 1) / T, T, 0, stream>>>(col, dinv);
    k_dinv     <<<(N_NODES + T - 1) / T, T, 0, stream>>>(dinv);

    // layer 1
    dim3 g1((M_TILES + WPB - 1) / WPB, D_H1 / 16);          // (391, 8)
    k_gemm_wmma<D_FEAT, D_H1><<<g1, T, 0, stream>>>(x, W1, h1);

    long long s1 = (long long)N_EDGES * (D_H1 / 4);
    k_scatter<D_H1><<<(unsigned)((s1 + T - 1) / T), T, 0, stream>>>(row, col, dinv, h1, agg1);

    long long p1 = (long long)N_NODES * D_H1;
    k_post<D_H1><<<(unsigned)((p1 + T - 1) / T), T, 0, stream>>>(h1, dinv, b1, agg1);

    // layer 2 (agg1 now holds y1; h1 region is dead -> reuse for h2/agg2)
    dim3 g2((M_TILES + WPB - 1) / WPB, D_H2 / 16);          // (391, 2)
    k_gemm_wmma<D_H1, D_H2><<<g2, T, 0, stream>>>(agg1, W2, h2);

    hipMemsetAsync(agg2, 0, (size_t)N_NODES * D_H2 * sizeof(float), stream);

    long long s2 = (long long)N_EDGES * (D_H2 / 4);
    k_scatter<D_H2><<<(unsigned)((s2 + T - 1) / T), T, 0, stream>>>(row, col, dinv, h2, agg2);

    long long p2 = (long long)N_NODES * D_H2;
    k_post<D_H2><<<(unsigned)((p2 + T - 1) / T), T, 0, stream>>>(h2, dinv, b2, agg2);

    // readout + softmax
    k_final<<<(N_NODES + T - 1) / T, T, 0, stream>>>(agg2, Wl, bl, out);
}